// Feature_Fields_2070174237336
// MI455X (gfx1250) — compile-verified
//
#include <hip/hip_runtime.h>
#include <hip/hip_bf16.h>

typedef float v2f __attribute__((ext_vector_type(2)));
typedef float v8f __attribute__((ext_vector_type(8)));

#define WD 768
#define KTOT 3072           // 4 neighbors * 768
#define NITER 192           // KTOT / 16
#define R2 0.25f
#define LN_EPS 1e-12f

// ---------------------------------------------------------------------------
// Kernel 1: top-4 nearest neighbors per query (ascending d2), radius mask.
// One 256-thread block per query. d2 = ||q||^2 + ||p||^2 - 2 q.p (matches ref).
// ---------------------------------------------------------------------------
__global__ __launch_bounds__(256) void knn_top4(const float* __restrict__ qpos,
                                                const float* __restrict__ pts,
                                                int M,
                                                int* __restrict__ idx_out,
                                                float* __restrict__ scl_out)
{
    __shared__ float sv[1024];
    __shared__ int   si[1024];
    const int n   = blockIdx.x;
    const int tid = threadIdx.x;

    const float qx = qpos[n * 3 + 0];
    const float qy = qpos[n * 3 + 1];
    const float qz = qpos[n * 3 + 2];
    const float qq = qx * qx + qy * qy + qz * qz;

    float bv0 = 3.4e38f, bv1 = 3.4e38f, bv2 = 3.4e38f, bv3 = 3.4e38f;
    int   bi0 = 0, bi1 = 0, bi2 = 0, bi3 = 0;

    for (int p = tid; p < M; p += 256) {
        const float px = pts[p * 3 + 0];
        const float py = pts[p * 3 + 1];
        const float pz = pts[p * 3 + 2];
        const float pp = px * px + py * py + pz * pz;
        const float dt = qx * px + qy * py + qz * pz;
        const float d2 = qq + pp - 2.0f * dt;
        if (d2 < bv3) {
            if (d2 < bv0)      { bv3=bv2; bi3=bi2; bv2=bv1; bi2=bi1; bv1=bv0; bi1=bi0; bv0=d2; bi0=p; }
            else if (d2 < bv1) { bv3=bv2; bi3=bi2; bv2=bv1; bi2=bi1; bv1=d2; bi1=p; }
            else if (d2 < bv2) { bv3=bv2; bi3=bi2; bv2=d2; bi2=p; }
            else               { bv3=d2; bi3=p; }
        }
    }
    sv[tid*4+0]=bv0; sv[tid*4+1]=bv1; sv[tid*4+2]=bv2; sv[tid*4+3]=bv3;
    si[tid*4+0]=bi0; si[tid*4+1]=bi1; si[tid*4+2]=bi2; si[tid*4+3]=bi3;
    __syncthreads();

    // tree merge of sorted 4-lists
    for (int s = 128; s >= 1; s >>= 1) {
        if (tid < s) {
            float av[4], cv[4], ov[4];
            int   ai[4], ci[4], oi[4];
            #pragma unroll
            for (int k = 0; k < 4; ++k) {
                av[k] = sv[tid*4 + k];       ai[k] = si[tid*4 + k];
                cv[k] = sv[(tid+s)*4 + k];   ci[k] = si[(tid+s)*4 + k];
            }
            int ia = 0, ic = 0;
            #pragma unroll
            for (int k = 0; k < 4; ++k) {
                const bool takeA = (ic >= 4) || ((ia < 4) && (av[ia] <= cv[ic]));
                if (takeA) { ov[k] = av[ia]; oi[k] = ai[ia]; ++ia; }
                else       { ov[k] = cv[ic]; oi[k] = ci[ic]; ++ic; }
            }
            #pragma unroll
            for (int k = 0; k < 4; ++k) { sv[tid*4 + k] = ov[k]; si[tid*4 + k] = oi[k]; }
        }
        __syncthreads();
    }

    if (tid < 4) {
        idx_out[n * 4 + tid] = si[tid];
        scl_out[n * 4 + tid] = (sv[tid] <= R2) ? 1.0f : 0.0f;
    }
}

// ---------------------------------------------------------------------------
// Kernel 2: fused gather-GEMM (fp32 WMMA 16x16x4, double-buffered LDS)
//           + LayerNorm(agg) + position embedding GEMV + LayerNorm(pos) + add.
// Block = 256 threads (8 waves), computes 16 full output rows (16 x 768).
// Wave w owns columns [96w, 96w+96): 6 accumulators of 16x16.
// B tile staged pair-swizzled; wave w stages K-pair row p == w.
// ---------------------------------------------------------------------------
__global__ __launch_bounds__(256) void fused_agg(
    const float* __restrict__ feats,
    const float* __restrict__ pos_in,
    const float* __restrict__ Wagg,
    const float* __restrict__ bagg,
    const float* __restrict__ gagg,
    const float* __restrict__ beagg,
    const float* __restrict__ Wpos,
    const float* __restrict__ bpos,
    const float* __restrict__ gpos,
    const float* __restrict__ bepos,
    const int*   __restrict__ idx4,
    const float* __restrict__ scl4,
    float*       __restrict__ out)
{
    extern __shared__ float smem[];
    // double-buffered pair-swizzled tiles:
    //   As2[buf][(kk>>1)*32 + m*2 + (kk&1)]          16x16  -> 2*256 floats
    //   Bs2[buf][(kk>>1)*1536 + c*2 + (kk&1)]        16x768 -> 2*12288 floats
    float* As2  = smem;                 // 512
    float* Bs2  = smem + 512;           // 24576
    float* rsum = Bs2 + 24576;          // 16 per-row agg sums
    float* rssq = rsum + 16;            // 16 per-row agg sum of squared devs
    float* psum = rssq + 16;            // 16 per-row pos sums
    float* pssq = psum + 16;            // 16 per-row pos sum of squares
    float* rscl = pssq + 16;            // 16*4 neighbor radius masks
    int*   ridx = (int*)(rscl + 64);    // 16*4 neighbor indices
    // total = 512 + 24576 + 64 + 64 + 64 = 25280 floats = 101120 bytes

    const int tid  = threadIdx.x;
    const int wave = tid >> 5;
    const int lane = tid & 31;
    const int hi   = lane >> 4;
    const int lo   = lane & 15;
    const int n0   = blockIdx.x * 16;
    const int col0 = wave * 96 + lo;

    if (tid < 64) { ridx[tid] = idx4[n0 * 4 + tid]; rscl[tid] = scl4[n0 * 4 + tid]; }
    if (tid < 16) { rsum[tid] = 0.f; rssq[tid] = 0.f; psum[tid] = 0.f; pssq[tid] = 0.f; }
    __syncthreads();

    // ---- position-embedding stats (single pass, E[x^2]-mean^2) ----
    {
        const int m  = tid >> 4;
        const int cb = tid & 15;
        const float p0 = pos_in[(n0 + m) * 6 + 0];
        const float p1 = pos_in[(n0 + m) * 6 + 1];
        const float p2 = pos_in[(n0 + m) * 6 + 2];
        const float p3 = pos_in[(n0 + m) * 6 + 3];
        const float p4 = pos_in[(n0 + m) * 6 + 4];
        const float p5 = pos_in[(n0 + m) * 6 + 5];
        float s = 0.f, sq = 0.f;
        for (int j = 0; j < 48; ++j) {
            const int c = cb + 16 * j;
            const float v = bpos[c] + p0 * Wpos[0 * WD + c] + p1 * Wpos[1 * WD + c]
                                    + p2 * Wpos[2 * WD + c] + p3 * Wpos[3 * WD + c]
                                    + p4 * Wpos[4 * WD + c] + p5 * Wpos[5 * WD + c];
            s += v; sq += v * v;
        }
        atomicAdd(&psum[m], s);
        atomicAdd(&pssq[m], sq);
    }

    // ---- main gather-GEMM: K = 3072 in tiles of 16, double-buffered ----
    const int am   = tid >> 4;                              // A-stage row
    const int akk  = tid & 15;                              // A-stage k within tile
    const int aoff = (akk >> 1) * 32 + am * 2 + (akk & 1);  // swizzled A slot
    // this wave stages global rows (k0 + 2*wave) and (k0 + 2*wave + 1)
    const float* gB = Wagg + (size_t)(2 * wave) * WD + 2 * lane;

    float4 rq[12];      // staged B pair-quads for next tile
    float  ra;          // staged A element for next tile

    // prologue: load tile 0 into registers
    {
        #pragma unroll
        for (int j = 0; j < 12; ++j) {
            const v2f r0 = *(const v2f*)(gB + j * 64);
            const v2f r1 = *(const v2f*)(gB + WD + j * 64);
            rq[j].x = r0.x; rq[j].y = r1.x; rq[j].z = r0.y; rq[j].w = r1.y;
        }
        const int   fid = ridx[am * 4 + 0];
        const float sc  = rscl[am * 4 + 0];
        ra = sc * feats[(size_t)fid * WD + akk];
    }

    v8f acc[6] = {};
    size_t kbase = 0;       // global float offset of current tile: k0 * WD
    int nb = 0, j0 = 0;     // neighbor slot / feature column base of NEXT tile

    for (int it = 0; it < NITER; ++it) {
        const int buf = it & 1;
        float* Asb = As2 + buf * 256;
        float* Bsb = Bs2 + buf * 12288;

        // commit staged tile to LDS (conflict-free b128 stores)
        #pragma unroll
        for (int j = 0; j < 12; ++j)
            *(float4*)(Bsb + wave * 1536 + j * 128 + lane * 4) = rq[j];
        Asb[aoff] = ra;
        __syncthreads();

        // issue global loads for the next tile (overlaps with WMMA below)
        if (it + 1 < NITER) {
            kbase += 16 * WD;
            j0 += 16; if (j0 == WD) { j0 = 0; ++nb; }
            const float* g = gB + kbase;
            #pragma unroll
            for (int j = 0; j < 12; ++j) {
                const v2f r0 = *(const v2f*)(g + j * 64);
                const v2f r1 = *(const v2f*)(g + WD + j * 64);
                rq[j].x = r0.x; rq[j].y = r1.x; rq[j].z = r0.y; rq[j].w = r1.y;
            }
            const int   fid = ridx[am * 4 + nb];
            const float sc  = rscl[am * 4 + nb];
            ra = sc * feats[(size_t)fid * WD + j0 + akk];
        }

        // 24 x v_wmma_f32_16x16x4_f32
        #pragma unroll
        for (int ks = 0; ks < 4; ++ks) {
            const int p = 2 * ks + hi;          // K-pair index for this lane half
            const v2f a = *(const v2f*)&Asb[p * 32 + lo * 2];
            #pragma unroll
            for (int t = 0; t < 6; ++t) {
                const v2f b = *(const v2f*)&Bsb[p * 1536 + (col0 + t * 16) * 2];
                acc[t] = __builtin_amdgcn_wmma_f32_16x16x4_f32(
                    false, a, false, b, (short)0, acc[t], false, false);
            }
        }
    }

    // ---- agg LayerNorm stats: pass 1 (row sums) ----
    const float inv = 1.0f / 768.0f;
    float part[8];
    #pragma unroll
    for (int r = 0; r < 8; ++r) part[r] = 0.f;
    #pragma unroll
    for (int t = 0; t < 6; ++t) {
        const float bb = bagg[col0 + t * 16];
        #pragma unroll
        for (int r = 0; r < 8; ++r) part[r] += acc[t][r] + bb;
    }
    #pragma unroll
    for (int mk = 1; mk <= 8; mk <<= 1) {
        #pragma unroll
        for (int r = 0; r < 8; ++r) part[r] += __shfl_xor(part[r], mk, 32);
    }
    if (lo == 0) {
        #pragma unroll
        for (int r = 0; r < 8; ++r) atomicAdd(&rsum[r + 8 * hi], part[r]);
    }
    __syncthreads();

    // ---- pass 2 (row sum of squared deviations) ----
    float mean[8];
    #pragma unroll
    for (int r = 0; r < 8; ++r) { mean[r] = rsum[r + 8 * hi] * inv; part[r] = 0.f; }
    #pragma unroll
    for (int t = 0; t < 6; ++t) {
        const float bb = bagg[col0 + t * 16];
        #pragma unroll
        for (int r = 0; r < 8; ++r) {
            const float d = acc[t][r] + bb - mean[r];
            part[r] += d * d;
        }
    }
    #pragma unroll
    for (int mk = 1; mk <= 8; mk <<= 1) {
        #pragma unroll
        for (int r = 0; r < 8; ++r) part[r] += __shfl_xor(part[r], mk, 32);
    }
    if (lo == 0) {
        #pragma unroll
        for (int r = 0; r < 8; ++r) atomicAdd(&rssq[r + 8 * hi], part[r]);
    }
    __syncthreads();

    // ---- finalize: LN(agg) + LN(pos), store ----
    float amu[8], astd[8], pmu[8], pstd[8], pr[8][6];
    #pragma unroll
    for (int r = 0; r < 8; ++r) {
        const int m = r + 8 * hi;
        amu[r]  = rsum[m] * inv;
        astd[r] = rsqrtf(rssq[m] * inv + LN_EPS);
        const float pm = psum[m] * inv;
        float pv = pssq[m] * inv - pm * pm;
        if (pv < 0.f) pv = 0.f;
        pmu[r]  = pm;
        pstd[r] = rsqrtf(pv + LN_EPS);
        #pragma unroll
        for (int i = 0; i < 6; ++i) pr[r][i] = pos_in[(n0 + m) * 6 + i];
    }
    #pragma unroll
    for (int t = 0; t < 6; ++t) {
        const int c = col0 + t * 16;
        const float bb = bagg[c],  gg = gagg[c], be = beagg[c];
        const float pb = bpos[c],  pg = gpos[c], pe = bepos[c];
        const float w0 = Wpos[0 * WD + c], w1 = Wpos[1 * WD + c], w2 = Wpos[2 * WD + c];
        const float w3 = Wpos[3 * WD + c], w4 = Wpos[4 * WD + c], w5 = Wpos[5 * WD + c];
        #pragma unroll
        for (int r = 0; r < 8; ++r) {
            const int m = r + 8 * hi;
            const float x  = acc[t][r] + bb;
            const float av = (x - amu[r]) * astd[r] * gg + be;
            const float pvv = pb + pr[r][0] * w0 + pr[r][1] * w1 + pr[r][2] * w2
                                 + pr[r][3] * w3 + pr[r][4] * w4 + pr[r][5] * w5;
            const float po = (pvv - pmu[r]) * pstd[r] * pg + pe;
            out[(size_t)(n0 + m) * WD + c] = av + po;
        }
    }
}

// ---------------------------------------------------------------------------
extern "C" void kernel_launch(void* const* d_in, const int* in_sizes, int n_in,
                              void* d_out, int out_size, void* d_ws, size_t ws_size,
                              hipStream_t stream) {
    (void)n_in; (void)out_size; (void)ws_size;
    const float* q_pos  = (const float*)d_in[0];
    const float* pts    = (const float*)d_in[1];
    const float* feats  = (const float*)d_in[2];
    const float* pos_in = (const float*)d_in[3];
    const float* Wagg   = (const float*)d_in[4];
    const float* bagg   = (const float*)d_in[5];
    const float* gagg   = (const float*)d_in[6];
    const float* beagg  = (const float*)d_in[7];
    const float* Wpos   = (const float*)d_in[8];
    const float* bpos   = (const float*)d_in[9];
    const float* gpos   = (const float*)d_in[10];
    const float* bepos  = (const float*)d_in[11];
    float* out = (float*)d_out;

    const int N = in_sizes[0] / 3;   // 4096
    const int M = in_sizes[1] / 3;   // 32768

    int*   idx4 = (int*)d_ws;
    float* scl4 = (float*)((char*)d_ws + (size_t)N * 4 * sizeof(int));

    knn_top4<<<N, 256, 0, stream>>>(q_pos, pts, M, idx4, scl4);

    const size_t shmem = (size_t)(512 + 24576 + 64 + 64 + 64) * sizeof(float); // 101120 B
    fused_agg<<<N / 16, 256, shmem, stream>>>(feats, pos_in, Wagg, bagg, gagg, beagg,
                                              Wpos, bpos, gpos, bepos, idx4, scl4, out);
}